// GraphSAGELayer_69758858822456
// MI455X (gfx1250) — compile-verified
//
#include <hip/hip_runtime.h>
#include <hip/hip_bf16.h>

typedef float v2f __attribute__((ext_vector_type(2)));
typedef float v8f __attribute__((ext_vector_type(8)));

#define FEAT  128
#define UNITS 128
#define KTOT  256       // 2*FEAT
#define KC    32        // K chunk staged in LDS
#define BM    128       // rows per block tile
#define AS_STRIDE 36    // 32 data + 4 pad: m*36 % 64 distinct for 16 rows -> no bank conflicts
#define BS_STRIDE 136   // 128 data + 8 pad: half-wave bank sets disjoint (2*136 % 64 == 16)

// ---------------------------------------------------------------------------
// Edge scatter: one wave32 per edge. Each lane handles 4 consecutive features.
// neigh[row, :] += val * node_states[col, :], deg[row] += val (lane 0).
// node_states (51 MB) and neigh (51 MB) both L2-resident on MI455X (192 MB L2).
// ---------------------------------------------------------------------------
__global__ __launch_bounds__(256) void sage_edge_scatter(
    const float* __restrict__ X,     // [N,128] node_states
    const int*   __restrict__ erow,  // [E]
    const int*   __restrict__ ecol,  // [E]
    const float* __restrict__ eval,  // [E]
    float*       __restrict__ deg,   // [N]
    float*       __restrict__ neigh, // [N,128]
    int nEdges)
{
    const int lane = threadIdx.x & 31;
    const int e = (int)((blockIdx.x * (unsigned)blockDim.x + threadIdx.x) >> 5);
    if (e >= nEdges) return;

    const int   r = erow[e];
    const int   c = ecol[e];
    const float v = eval[e];

    if (lane == 0) {
        __hip_atomic_fetch_add(&deg[r], v, __ATOMIC_RELAXED, __HIP_MEMORY_SCOPE_AGENT);
    }

    const float4 x = ((const float4*)(X + (size_t)c * FEAT))[lane];
    float* dst = neigh + (size_t)r * FEAT + lane * 4;
    __hip_atomic_fetch_add(dst + 0, v * x.x, __ATOMIC_RELAXED, __HIP_MEMORY_SCOPE_AGENT);
    __hip_atomic_fetch_add(dst + 1, v * x.y, __ATOMIC_RELAXED, __HIP_MEMORY_SCOPE_AGENT);
    __hip_atomic_fetch_add(dst + 2, v * x.z, __ATOMIC_RELAXED, __HIP_MEMORY_SCOPE_AGENT);
    __hip_atomic_fetch_add(dst + 3, v * x.w, __ATOMIC_RELAXED, __HIP_MEMORY_SCOPE_AGENT);
}

// ---------------------------------------------------------------------------
// Tiled fused GEMM: out = relu([X | neigh*deg_inv] @ W + bias)
// Block tile 128x128, K chunked by 32 through LDS, V_WMMA_F32_16X16X4_F32.
//
// 8 waves: wave w owns rows [w*16, +16) of the block tile and sweeps all
// 8 column tiles with 8 independent accumulator chains (A fragment reused).
//
// WMMA f32 16x16x4 layouts (ISA 7.12.2):
//   A: lane L -> M = L%16, VGPRs hold K = 2*(L/16), 2*(L/16)+1
//   B: lane L -> N = L%16, rows  K = 2*(L/16), 2*(L/16)+1
//   C/D: VGPR r, lane L -> M = r + 8*(L/16), N = L%16
// ---------------------------------------------------------------------------
__global__ __launch_bounds__(256) void sage_wmma_gemm(
    const float* __restrict__ X,     // [N,128]
    const float* __restrict__ W,     // [256,128]
    const float* __restrict__ bias,  // [128]
    const float* __restrict__ deg,   // [N]
    const float* __restrict__ neigh, // [N,128]
    float*       __restrict__ out,   // [N,128]
    int nNodes)
{
    __shared__ float As[BM * AS_STRIDE];     // 18.0 KB  [m][k] padded
    __shared__ float Bs[KC * BS_STRIDE];     // 17.0 KB  [k][n] padded
    __shared__ float dinvS[BM];              //  0.5 KB

    const int tid  = threadIdx.x;
    const int wave = tid >> 5;               // 0..7 -> row sub-tile of block
    const int lane = tid & 31;
    const int mn   = lane & 15;              // M for A frag, N for B frag
    const int kh   = (lane >> 4) * 2;        // K pair selector: 0 or 2
    const int row0 = blockIdx.x * BM;

    // Per-row 1/deg (reciprocal_no_nan) staged once per block.
    if (tid < BM) {
        int rg = row0 + tid;
        if (rg >= nNodes) rg = nNodes - 1;
        const float d = deg[rg];
        dinvS[tid] = (d != 0.0f) ? (1.0f / d) : 0.0f;
    }

    v8f acc[8];
    #pragma unroll
    for (int ct = 0; ct < 8; ++ct) acc[ct] = (v8f){};

    // K chunks: 0..3 read X (scale 1), 4..7 read neigh (scale 1/deg).
    for (int kc = 0; kc < KTOT / KC; ++kc) {
        __syncthreads();   // protect LDS from previous chunk's readers

        const bool selfHalf = (kc < 4);
        const float* __restrict__ src = selfHalf ? X : neigh;
        const int koff = (kc & 3) * KC;      // offset within [0,128) feature block

        // Stage A chunk: 128 rows x 32 K = 1024 float4s, 4 per thread.
        #pragma unroll
        for (int i = tid; i < (BM * KC) / 4; i += 256) {
            const int m = i >> 3;            // 8 float4 per row
            const int k = (i & 7) * 4;
            int rg = row0 + m;
            if (rg >= nNodes) rg = nNodes - 1;
            float4 v = *(const float4*)(src + (size_t)rg * FEAT + koff + k);
            if (!selfHalf) {
                const float s = dinvS[m];
                v.x *= s; v.y *= s; v.z *= s; v.w *= s;
            }
            *(float4*)(&As[m * AS_STRIDE + k]) = v;
        }

        // Stage B chunk: 32 K x 128 N = 1024 float4s, 4 per thread.
        #pragma unroll
        for (int i = tid; i < (KC * UNITS) / 4; i += 256) {
            const int k  = i >> 5;           // 32 float4 per K row
            const int n4 = (i & 31) * 4;
            float4 v = *(const float4*)(W + (size_t)(kc * KC + k) * UNITS + n4);
            *(float4*)(&Bs[k * BS_STRIDE + n4]) = v;
        }

        __syncthreads();

        // Compute: 8 independent WMMA chains (one per column tile).
        const int m0 = wave * 16;
        #pragma unroll
        for (int k = 0; k < KC; k += 4) {
            v2f a;
            a.x = As[(m0 + mn) * AS_STRIDE + k + kh];
            a.y = As[(m0 + mn) * AS_STRIDE + k + kh + 1];
            #pragma unroll
            for (int ct = 0; ct < 8; ++ct) {
                v2f b;
                b.x = Bs[(k + kh)     * BS_STRIDE + ct * 16 + mn];
                b.y = Bs[(k + kh + 1) * BS_STRIDE + ct * 16 + mn];
                acc[ct] = __builtin_amdgcn_wmma_f32_16x16x4_f32(
                              false, a, false, b, (short)0, acc[ct], false, false);
            }
        }
    }

    // Bias + ReLU + predicated stores (WMMAs all done; divergence is safe now).
    const int mbase = (lane >> 4) * 8;       // D layout: M = r + 8*(lane/16)
    #pragma unroll
    for (int ct = 0; ct < 8; ++ct) {
        const int   col  = ct * 16 + mn;
        const float bcol = bias[col];
        #pragma unroll
        for (int r = 0; r < 8; ++r) {
            const int rowg = row0 + wave * 16 + mbase + r;
            if (rowg < nNodes) {
                float v = acc[ct][r] + bcol;
                v = v > 0.0f ? v : 0.0f;
                out[(size_t)rowg * UNITS + col] = v;
            }
        }
    }
}

// ---------------------------------------------------------------------------
// Launch: inputs are (node_states, edge_row, edge_col, edge_val, kernel, bias).
// Workspace: deg[N] floats at offset 0, neigh[N*128] floats after it.
// ---------------------------------------------------------------------------
extern "C" void kernel_launch(void* const* d_in, const int* in_sizes, int n_in,
                              void* d_out, int out_size, void* d_ws, size_t ws_size,
                              hipStream_t stream) {
    const float* X    = (const float*)d_in[0];
    const int*   erow = (const int*)  d_in[1];
    const int*   ecol = (const int*)  d_in[2];
    const float* eval = (const float*)d_in[3];
    const float* W    = (const float*)d_in[4];
    const float* bias = (const float*)d_in[5];

    const int N = in_sizes[0] / FEAT;
    const int E = in_sizes[1];

    float* deg   = (float*)d_ws;
    float* neigh = deg + N;                  // N*4 bytes, 16B aligned for N%4==0

    // Zero degree + neighbor-sum accumulators (graph-capture-safe memset).
    hipMemsetAsync(d_ws, 0, (size_t)(N + (size_t)N * FEAT) * sizeof(float), stream);

    // Edge scatter: one wave32 per edge -> E*32 threads.
    {
        const long long threads = (long long)E * 32;
        const int blk = 256;
        const int grid = (int)((threads + blk - 1) / blk);
        sage_edge_scatter<<<grid, blk, 0, stream>>>(X, erow, ecol, eval, deg, neigh, E);
    }

    // Tiled fused normalize + GEMM + bias + ReLU: 128 rows per block.
    {
        const int grid = (N + BM - 1) / BM;
        sage_wmma_gemm<<<grid, 256, 0, stream>>>(X, W, bias, deg, neigh, (float*)d_out, N);
    }
}